// GNNFusion_67791763800785
// MI455X (gfx1250) — compile-verified
//
#include <hip/hip_runtime.h>
#include <hip/hip_bf16.h>

typedef __attribute__((ext_vector_type(2))) float v2f;
typedef __attribute__((ext_vector_type(8))) float v8f;

#define NEG_SLOPE 0.01f

__device__ __forceinline__ float lrelu(float x) {
    return x >= 0.0f ? x : NEG_SLOPE * x;
}

// ---------------------------------------------------------------------------
// Degree / normalization kernels
// ---------------------------------------------------------------------------
__global__ void deg_init_kernel(float* deg, int N) {
    int i = blockIdx.x * blockDim.x + threadIdx.x;
    if (i < N) deg[i] = 1.0f;  // self-loop contribution
}

__global__ void deg_count_kernel(const long long* __restrict__ col, float* deg, int E) {
    int i = blockIdx.x * blockDim.x + threadIdx.x;
    if (i < E) atomicAdd(&deg[(int)col[i]], 1.0f);
}

__global__ void deg_rsqrt_kernel(float* deg, int N) {
    int i = blockIdx.x * blockDim.x + threadIdx.x;
    if (i < N) deg[i] = rsqrtf(deg[i]);  // deg >= 1 always (self loop)
}

// ---------------------------------------------------------------------------
// WMMA fp32 GEMM: C[row, coff+n] = A[M,K] @ W[K,NC]  (+bias, +leaky if MODE==1)
// block = 256 threads (8 waves); block computes a 16 x 128 output slab.
// Uses V_WMMA_F32_16X16X4_F32: A 16x4 (2 VGPR), B 4x16 (2 VGPR), C/D 16x16 (8 VGPR)
// ---------------------------------------------------------------------------
template <int MODE>
__global__ void gemm_wmma_f32(const float* __restrict__ A, const float* __restrict__ W,
                              const float* __restrict__ bias, float* __restrict__ C,
                              int M, int K, int NC, int ldc, int coff) {
    __shared__ float As[16 * 256];  // 16 rows x up to K=256 cols (16 KB of the 320 KB LDS)

    const int mbase = blockIdx.x * 16;

    // Cooperative, coalesced load of the 16 x K A-tile into LDS (zero-pad OOB rows)
    for (int r = 0; r < 16; ++r) {
        const int grow = mbase + r;
        for (int c = threadIdx.x; c < K; c += blockDim.x) {
            As[r * K + c] = (grow < M) ? A[(size_t)grow * K + c] : 0.0f;
        }
    }
    __syncthreads();

    const int lane = threadIdx.x & 31;
    const int wave = threadIdx.x >> 5;
    const int colbase = (blockIdx.y * 8 + wave) * 16;
    if (colbase >= NC) return;  // wave-uniform

    const int mr = lane & 15;          // A row (lanes 0-15 / 16-31 both map M=0..15)
    const int kb = (lane >> 4) << 1;   // K sub-offset within the 4-wide step: 0 or 2

    v8f acc = {};
    for (int k0 = 0; k0 < K; k0 += 4) {
        v2f a, b;
        // A 16x4 layout: lane (l&15)=M, VGPR0/1 = K = kb, kb+1
        a.x = As[mr * K + k0 + kb];
        a.y = As[mr * K + k0 + kb + 1];
        // B 4x16 layout: lane (l&15)=N, VGPR0/1 = K = kb, kb+1
        const float* wp = W + (size_t)(k0 + kb) * NC + colbase + mr;
        b.x = wp[0];
        b.y = wp[NC];
        // 8 args: (neg_a, A, neg_b, B, c_mod, C, reuse_a, reuse_b)
        acc = __builtin_amdgcn_wmma_f32_16x16x4_f32(
            false, a, false, b, (short)0, acc, false, false);
    }

    // C/D 16x16 layout: VGPR i -> M = i + 8*(lane>=16), N = lane&15
    const int n = colbase + (lane & 15);
    const int rb = mbase + ((lane >> 4) << 3);
#pragma unroll
    for (int i = 0; i < 8; ++i) {
        const int row = rb + i;
        if (row < M) {
            float v = acc[i];
            if (MODE == 1) { v += bias[n]; v = lrelu(v); }
            C[(size_t)row * ldc + coff + n] = v;
        }
    }
}

// ---------------------------------------------------------------------------
// agg[i, :] = h[i, :] * dinv[i]^2   (self-loop term initializes the accumulator)
// ---------------------------------------------------------------------------
__global__ void selfloop_init_kernel(const float* __restrict__ h, const float* __restrict__ dinv,
                                     float* __restrict__ agg, int total) {
    int i = blockIdx.x * blockDim.x + threadIdx.x;
    if (i < total) {
        float d = dinv[i >> 7];
        agg[i] = h[i] * d * d;
    }
}

// ---------------------------------------------------------------------------
// Edge scatter: one wave per edge; each lane handles 4 of the 128 features.
// Gathers are 512 B coalesced; h (51 MB) is L2-resident on the 192 MB L2.
// ---------------------------------------------------------------------------
__global__ void edge_scatter_kernel(const long long* __restrict__ row, const long long* __restrict__ col,
                                    const float* __restrict__ dinv, const float* __restrict__ h,
                                    float* __restrict__ agg, int E) {
    int e = blockIdx.x * (blockDim.x >> 5) + (threadIdx.x >> 5);
    if (e >= E) return;
    const int r = (int)row[e];
    const int c = (int)col[e];
    const float nrm = dinv[r] * dinv[c];
    const int lane = threadIdx.x & 31;
    const float4 hv = ((const float4*)(h + (size_t)r * 128))[lane];
    float* dst = agg + (size_t)c * 128 + lane * 4;
    atomicAdd(dst + 0, hv.x * nrm);
    atomicAdd(dst + 1, hv.y * nrm);
    atomicAdd(dst + 2, hv.z * nrm);
    atomicAdd(dst + 3, hv.w * nrm);
}

// out[i] = leaky(agg[i] + bias[i % 128])
__global__ void bias_leaky_kernel(const float* __restrict__ agg, const float* __restrict__ bias,
                                  float* __restrict__ out, int total) {
    int i = blockIdx.x * blockDim.x + threadIdx.x;
    if (i < total) out[i] = lrelu(agg[i] + bias[i & 127]);
}

__global__ void zero_kernel(float* p, int n) {
    int i = blockIdx.x * blockDim.x + threadIdx.x;
    if (i < n) p[i] = 0.0f;
}

// One wave per node: pooled[batch[i]] += h2[i]; cnt[batch[i]] += 1
__global__ void pool_sum_kernel(const float* __restrict__ h2, const long long* __restrict__ batch,
                                float* __restrict__ pooled, float* __restrict__ cnt, int N) {
    int node = blockIdx.x * (blockDim.x >> 5) + (threadIdx.x >> 5);
    if (node >= N) return;
    const int g = (int)batch[node];
    const int lane = threadIdx.x & 31;
    const float4 hv = ((const float4*)(h2 + (size_t)node * 128))[lane];
    float* dst = pooled + (size_t)g * 128 + lane * 4;
    atomicAdd(dst + 0, hv.x);
    atomicAdd(dst + 1, hv.y);
    atomicAdd(dst + 2, hv.z);
    atomicAdd(dst + 3, hv.w);
    if (lane == 0) atomicAdd(&cnt[g], 1.0f);
}

__global__ void pool_div_kernel(float* pooled, const float* __restrict__ cnt, int total) {
    int i = blockIdx.x * blockDim.x + threadIdx.x;
    if (i < total) pooled[i] = pooled[i] / fmaxf(cnt[i >> 7], 1.0f);
}

// out[g] = dot(z1[g, :128], Wm2[:,0]) + bm2 ; one wave per graph, shuffle reduce
__global__ void final_dot_kernel(const float* __restrict__ z1, const float* __restrict__ Wm2,
                                 const float* __restrict__ bm2, float* __restrict__ out, int G) {
    int g = blockIdx.x * (blockDim.x >> 5) + (threadIdx.x >> 5);
    if (g >= G) return;
    const int lane = threadIdx.x & 31;
    const float4 zv = ((const float4*)(z1 + (size_t)g * 128))[lane];
    const float4 wv = ((const float4*)Wm2)[lane];
    float s = zv.x * wv.x + zv.y * wv.y + zv.z * wv.z + zv.w * wv.w;
#pragma unroll
    for (int off = 16; off > 0; off >>= 1) s += __shfl_xor(s, off, 32);
    if (lane == 0) out[g] = s + bm2[0];
}

// ---------------------------------------------------------------------------
// Host-side launch sequence
// ---------------------------------------------------------------------------
extern "C" void kernel_launch(void* const* d_in, const int* in_sizes, int n_in,
                              void* d_out, int out_size, void* d_ws, size_t ws_size,
                              hipStream_t stream) {
    (void)n_in; (void)out_size; (void)ws_size;

    const float*     x     = (const float*)d_in[0];
    const long long* ei    = (const long long*)d_in[1];
    const long long* batch = (const long long*)d_in[2];
    const float*     gfeat = (const float*)d_in[3];
    const float *W1 = (const float*)d_in[4],  *b1  = (const float*)d_in[5];
    const float *W2 = (const float*)d_in[6],  *b2  = (const float*)d_in[7];
    const float *Wg = (const float*)d_in[8],  *bg  = (const float*)d_in[9];
    const float *Wf = (const float*)d_in[10], *bf  = (const float*)d_in[11];
    const float *Wm1= (const float*)d_in[12], *bm1 = (const float*)d_in[13];
    const float *Wm2= (const float*)d_in[14], *bm2 = (const float*)d_in[15];
    float* out = (float*)d_out;

    const int N = in_sizes[0] / 64;   // 100000
    const int E = in_sizes[1] / 2;    // 1600000
    const int G = in_sizes[3] / 32;   // 256
    const long long* row = ei;
    const long long* col = ei + E;

    // Workspace layout (floats), 16-byte aligned chunks
    float* ws = (float*)d_ws;
    const size_t No = ((size_t)N + 3) & ~(size_t)3;
    float* dinv   = ws;                         // N      (deg -> dinv in place)
    float* T      = dinv + No;                  // N*128
    float* U      = T + (size_t)N * 128;        // N*128
    float* pooled = U + (size_t)N * 128;        // G*128
    float* cnt    = pooled + (size_t)G * 128;   // G
    float* fused  = cnt + (((size_t)G + 3) & ~(size_t)3);  // G*256
    float* z1     = fused + (size_t)G * 256;    // G*128

    const int NH = N * 128;
    const int TPB = 256;

    // --- symmetric normalization: dinv = rsqrt(in_deg + 1) --------------------
    deg_init_kernel<<<(N + TPB - 1) / TPB, TPB, 0, stream>>>(dinv, N);
    deg_count_kernel<<<(E + TPB - 1) / TPB, TPB, 0, stream>>>(col, dinv, E);
    deg_rsqrt_kernel<<<(N + TPB - 1) / TPB, TPB, 0, stream>>>(dinv, N);

    // --- GCN layer 1 ----------------------------------------------------------
    gemm_wmma_f32<0><<<dim3((N + 15) / 16, 1), TPB, 0, stream>>>(x, W1, nullptr, T, N, 64, 128, 128, 0);
    selfloop_init_kernel<<<(NH + TPB - 1) / TPB, TPB, 0, stream>>>(T, dinv, U, NH);
    edge_scatter_kernel<<<(E + 7) / 8, TPB, 0, stream>>>(row, col, dinv, T, U, E);
    bias_leaky_kernel<<<(NH + TPB - 1) / TPB, TPB, 0, stream>>>(U, b1, T, NH);   // T = h1

    // --- GCN layer 2 ----------------------------------------------------------
    gemm_wmma_f32<0><<<dim3((N + 15) / 16, 1), TPB, 0, stream>>>(T, W2, nullptr, U, N, 128, 128, 128, 0);
    selfloop_init_kernel<<<(NH + TPB - 1) / TPB, TPB, 0, stream>>>(U, dinv, T, NH);
    edge_scatter_kernel<<<(E + 7) / 8, TPB, 0, stream>>>(row, col, dinv, U, T, E);
    bias_leaky_kernel<<<(NH + TPB - 1) / TPB, TPB, 0, stream>>>(T, b2, U, NH);   // U = h2

    // --- global mean pool -----------------------------------------------------
    zero_kernel<<<(G * 128 + TPB - 1) / TPB, TPB, 0, stream>>>(pooled, G * 128);
    zero_kernel<<<(G + TPB - 1) / TPB, TPB, 0, stream>>>(cnt, G);
    pool_sum_kernel<<<(N + 7) / 8, TPB, 0, stream>>>(U, batch, pooled, cnt, N);
    pool_div_kernel<<<(G * 128 + TPB - 1) / TPB, TPB, 0, stream>>>(pooled, cnt, G * 128);

    // --- fusion MLP (leaky folded into GEMM stores) ---------------------------
    // fused[:, 0:128]   = leaky(pooled @ Wg + bg)
    gemm_wmma_f32<1><<<dim3((G + 15) / 16, 1), TPB, 0, stream>>>(pooled, Wg, bg, fused, G, 128, 128, 256, 0);
    // fused[:, 128:256] = leaky(gfeat @ Wf + bf)
    gemm_wmma_f32<1><<<dim3((G + 15) / 16, 1), TPB, 0, stream>>>(gfeat, Wf, bf, fused, G, 32, 128, 256, 128);
    // z1 = leaky(fused @ Wm1 + bm1)
    gemm_wmma_f32<1><<<dim3((G + 15) / 16, 1), TPB, 0, stream>>>(fused, Wm1, bm1, z1, G, 256, 128, 128, 0);
    // out = z1 @ Wm2 + bm2
    final_dot_kernel<<<(G + 7) / 8, TPB, 0, stream>>>(z1, Wm2, bm2, out, G);
}